// BiTemperedLoss_55997783605590
// MI455X (gfx1250) — compile-verified
//
#include <hip/hip_runtime.h>

// ---------------------------------------------------------------------------
// Bi-tempered logistic loss, t1=0.8, t2=1.3, label_smoothing=0.2, 5 iters.
// inputs/targets: (32,4,512,512) f32.  Rows = channel dim (4 rows x 8.4M).
// 8 full passes total (1 max, 6 Z-sums, 1 loss). Input (134MB) fits in 192MB
// L2 -> recompute sigmoid per pass instead of spilling a0.
// Block reductions use V_WMMA_F32_16X16X4_F32 (ones-vector matmul reduce).
// ---------------------------------------------------------------------------

typedef __attribute__((ext_vector_type(2))) float v2f;
typedef __attribute__((ext_vector_type(8))) float v8f;

#define TPB      256
#define EPB      16384            // elements per block (contiguous)
#define NBLK     2048             // 33,554,432 / 16384
#define VPT      16               // float4 loads per thread

// t2 = 1.3: 1-t2 = -0.3, 1/(1-t2) = -10/3
#define NEG_INV_OMT2  (-3.33333325f)
// t1 = 0.8: log_t(u,0.8) = (u^0.2 - 1)/0.2 ; 2-t1 = 1.2
// label smoothing, K = 8388608 (computed in double, folded to float)
#define LS_C1  (0.7999999761581421f)    // 1 - K/(K-1)*0.2
#define LS_C0  (2.3841860752327195e-8f) // 0.2/(K-1)

__device__ __forceinline__ float fsigmoid(float v) {
    float e = __builtin_amdgcn_exp2f(-v * 1.4426950408889634f);
    return __builtin_amdgcn_rcpf(1.0f + e);
}
__device__ __forceinline__ float flog2(float v) { return __builtin_amdgcn_logf(v); }
__device__ __forceinline__ float fexp2(float v) { return __builtin_amdgcn_exp2f(v); }

// Deterministic block-wide sum of 256 values via V_WMMA_F32_16X16X4_F32.
// Stage partials in LDS; wave 0 does 4 WMMAs (A = 16x4 data tile, B = ones)
// accumulating into one 16x16 f32 C, then one xor-16 shuffle finishes.
__device__ __forceinline__ void block_sum_store(float v, float* red, float* dst) {
    const int t = threadIdx.x;
    red[t] = v;
    __syncthreads();
    if (t < 32) {                       // wave 0, EXEC all ones within the wave
        const int r  = t & 15;
        const int kk = (t >> 4) << 1;   // lanes 0-15 -> K=0,1 ; lanes 16-31 -> K=2,3
        v2f ones; ones.x = 1.0f; ones.y = 1.0f;
        v8f c = {};
#pragma unroll
        for (int ch = 0; ch < 4; ++ch) {
            const float* b = red + ch * 64 + r * 4 + kk;  // A(m,k) = red[ch*64 + m*4 + k]
            v2f a; a.x = b[0]; a.y = b[1];
            c = __builtin_amdgcn_wmma_f32_16x16x4_f32(
                    false, a, false, ones, (short)0, c, false, false);
        }
        // lane n<16 holds rows 0..7 of col n; lane n>=16 rows 8..15 of col n-16
        float s = c[0] + c[1] + c[2] + c[3] + c[4] + c[5] + c[6] + c[7];
        s += __shfl_xor(s, 16, 32);
        if (t == 0) *dst = s;
    }
}

// ---- Pass 1: per-block max of raw inputs (sigmoid monotone) ----------------
__global__ void kmax(const float4* __restrict__ in, float* __restrict__ part) {
    const int t = threadIdx.x;
    const int base = blockIdx.x * (EPB / 4);
    float m = -3.4e38f;
#pragma unroll 4
    for (int i = 0; i < VPT; ++i) {
        float4 v = in[base + i * TPB + t];
        m = fmaxf(m, fmaxf(fmaxf(v.x, v.y), fmaxf(v.z, v.w)));
    }
    __shared__ float red[TPB];
    red[t] = m;
    __syncthreads();
    for (int s = TPB / 2; s > 0; s >>= 1) {
        if (t < s) red[t] = fmaxf(red[t], red[t + s]);
        __syncthreads();
    }
    if (t == 0) part[blockIdx.x] = red[0];
}

// ---- mu per row; init s = 1. 4 waves, one per row. -------------------------
__global__ void kmu(const float* __restrict__ part, float* __restrict__ scal) {
    const int t = threadIdx.x, w = t >> 5, l = t & 31;
    float m = -3.4e38f;
    const int base = (((l << 2) + w) << 4);       // blocks of row w, group l
#pragma unroll
    for (int j = 0; j < 16; ++j) m = fmaxf(m, part[base + j]);
    for (int off = 16; off > 0; off >>= 1) m = fmaxf(m, __shfl_xor(m, off, 32));
    if (l == 0) {
        scal[w]     = fsigmoid(m);   // mu (in sigmoid space)
        scal[4 + w] = 1.0f;          // s0 = 1
    }
}

// ---- Z-sum pass: acc += (1 - 0.3*s*(x - mu))^(-10/3) -----------------------
__global__ void ksum(const float4* __restrict__ in, const float* __restrict__ scal,
                     float* __restrict__ part) {
    const int t = threadIdx.x;
    const int c = (blockIdx.x >> 4) & 3;
    const float mu = scal[c];
    const float k  = 0.3f * scal[4 + c];
    const float b0 = 1.0f + k * mu;               // b = b0 - k*x
    const int base = blockIdx.x * (EPB / 4);
    float acc = 0.0f;
#pragma unroll 4
    for (int i = 0; i < VPT; ++i) {
        float4 v = in[base + i * TPB + t];
        acc += fexp2(NEG_INV_OMT2 * flog2(b0 - k * fsigmoid(v.x)));
        acc += fexp2(NEG_INV_OMT2 * flog2(b0 - k * fsigmoid(v.y)));
        acc += fexp2(NEG_INV_OMT2 * flog2(b0 - k * fsigmoid(v.z)));
        acc += fexp2(NEG_INV_OMT2 * flog2(b0 - k * fsigmoid(v.w)));
    }
    __shared__ float red[TPB];
    block_sum_store(acc, red, &part[blockIdx.x]);
}

// ---- Z reduce + update: iter -> s = Z^-0.3 ; final -> offs = (Z^0.3-1)/0.3 -
__global__ void kupd(const float* __restrict__ part, float* __restrict__ scal,
                     int final_pass) {
    const int t = threadIdx.x, w = t >> 5, l = t & 31;
    float z = 0.0f;
    const int base = (((l << 2) + w) << 4);
#pragma unroll
    for (int j = 0; j < 16; ++j) z += part[base + j];
    for (int off = 16; off > 0; off >>= 1) z += __shfl_xor(z, off, 32);
    if (l == 0) {
        const float lz = flog2(z);
        if (final_pass) scal[8 + w] = (fexp2(0.3f * lz) - 1.0f) * (1.0f / 0.3f);
        else            scal[4 + w] = fexp2(-0.3f * lz);
    }
}

// ---- Loss pass -------------------------------------------------------------
__global__ void kloss(const float4* __restrict__ in, const float4* __restrict__ tg,
                      const float* __restrict__ scal, float* __restrict__ part) {
    const int t = threadIdx.x;
    const int c = (blockIdx.x >> 4) & 3;
    const float mu   = scal[c];
    const float offs = scal[8 + c];               // norm - mu
    const float b0   = 1.0f + 0.3f * (mu + offs); // b = b0 - 0.3*x, x=sigmoid
    const int base = blockIdx.x * (EPB / 4);
    float acc = 0.0f;
#pragma unroll 2
    for (int i = 0; i < VPT; ++i) {
        float4 v = in[base + i * TPB + t];
        float4 u = tg[base + i * TPB + t];
        float xs[4] = {v.x, v.y, v.z, v.w};
        float ys[4] = {u.x, u.y, u.z, u.w};
#pragma unroll
        for (int j = 0; j < 4; ++j) {
            float x = fsigmoid(xs[j]);
            float q = NEG_INV_OMT2 * flog2(b0 - 0.3f * x);   // log2(p)
            float y = LS_C1 * ys[j] + LS_C0;                 // smoothed label
            float ly  = flog2(y + 1e-10f);
            float y02 = fexp2(0.2f * ly);                    // (y+eps)^0.2
            float y12 = fexp2(1.2f * flog2(y));              // y^1.2
            float p02 = fexp2(0.2f * q);                     // p^0.2
            float p12 = fexp2(1.2f * q);                     // p^1.2
            // temp1 = 5*(y02 - p02)*y ; temp2 = (y12 - p12)/1.2
            acc += 5.0f * (y02 - p02) * y + (1.0f / 1.2f) * (y12 - p12);
        }
    }
    __shared__ float red[TPB];
    block_sum_store(acc, red, &part[blockIdx.x]);
}

// ---- Final: 4 row sums -> mean ---------------------------------------------
__global__ void kfin(const float* __restrict__ part, float* __restrict__ out) {
    const int t = threadIdx.x, w = t >> 5, l = t & 31;
    float z = 0.0f;
    const int base = (((l << 2) + w) << 4);
#pragma unroll
    for (int j = 0; j < 16; ++j) z += part[base + j];
    for (int off = 16; off > 0; off >>= 1) z += __shfl_xor(z, off, 32);
    __shared__ float rl[4];
    if (l == 0) rl[w] = z;
    __syncthreads();
    if (t == 0) out[0] = 0.25f * (rl[0] + rl[1] + rl[2] + rl[3]);
}

extern "C" void kernel_launch(void* const* d_in, const int* in_sizes, int n_in,
                              void* d_out, int out_size, void* d_ws, size_t ws_size,
                              hipStream_t stream) {
    (void)in_sizes; (void)n_in; (void)out_size; (void)ws_size;
    const float4* in = (const float4*)d_in[0];
    const float4* tg = (const float4*)d_in[1];
    float* wsf  = (float*)d_ws;
    float* part = wsf;            // 2048 block partials
    float* scal = wsf + NBLK;     // [0..3]=mu  [4..7]=s  [8..11]=offs

    kmax<<<NBLK, TPB, 0, stream>>>((const float4*)in, part);
    kmu <<<1, 128, 0, stream>>>(part, scal);
    for (int it = 0; it < 6; ++it) {                 // 5 fixed-point + 1 final Z
        ksum<<<NBLK, TPB, 0, stream>>>(in, scal, part);
        kupd<<<1, 128, 0, stream>>>(part, scal, (it == 5) ? 1 : 0);
    }
    kloss<<<NBLK, TPB, 0, stream>>>(in, tg, scal, part);
    kfin <<<1, 128, 0, stream>>>(part, (float*)d_out);
}